// BDC_54631984005383
// MI455X (gfx1250) — compile-verified
//
#include <hip/hip_runtime.h>
#include <math.h>

#define DIM 640
#define MSP 100
#define NB  64
#define EPS 1e-5f
#define NT  40                        // DIM / 16 tile blocks per side
#define TRIU_PER_B (DIM*(DIM+1)/2)    // 205120
#define INV_DIM (1.0f/640.0f)
#define BLOCKS_PER_B 205              // 820 triu tiles / 4 waves per block

typedef float v2f __attribute__((ext_vector_type(2)));
typedef float v8f __attribute__((ext_vector_type(8)));

// ---------------------------------------------------------------------------
// Map linear triu-tile index (0..819) -> (I, J), J >= I. Wave-uniform.
// ---------------------------------------------------------------------------
__device__ inline void decode_triu(int tIdx, int& I, int& J) {
    int rem = tIdx; I = 0;
    while (rem >= NT - I) { rem -= NT - I; ++I; }
    J = I + rem;
}

// ---------------------------------------------------------------------------
// Stage one 16x100 f32 tile (contiguous 6.4 KB in memory) into LDS,
// coalesced float4 loads: 400 float4s across 32 lanes.
// ---------------------------------------------------------------------------
__device__ inline void stage_tile(const float* __restrict__ src,
                                  float* dst, int lane) {
    const float4* s4 = (const float4*)src;
    float4* d4 = (float4*)dst;
    #pragma unroll
    for (int i = 0; i < 13; ++i) {
        int idx = lane + i * 32;
        if (idx < 400) d4[idx] = s4[idx];
    }
}

// ---------------------------------------------------------------------------
// 16x16 Gram tile from LDS-resident A/B row blocks, K=100 via 25 fp32 WMMAs.
// A-tile 16x4 f32: lane(h=lane>>4, m=lane&15) holds K = k0+2h+{0,1} of row m;
// B-tile 4x16 for x*x^T has the identical per-lane indexing on the J rows.
// C/D: VGPR r -> row (r + 8h), col (lane&15). LDS reads are conflict-free.
// ---------------------------------------------------------------------------
__device__ inline v8f gram_tile_lds(const float* a, const float* bb,
                                    int m16, int h) {
    const float* pa = a  + m16 * MSP + 2 * h;
    const float* pb = bb + m16 * MSP + 2 * h;
    v8f c = {};
    #pragma unroll
    for (int kt = 0; kt < 25; ++kt) {
        v2f av = *(const v2f*)(pa + kt * 4);
        v2f bv = *(const v2f*)(pb + kt * 4);
        c = __builtin_amdgcn_wmma_f32_16x16x4_f32(
                false, av, false, bv, (short)0, c, false, false);
    }
    return c;
}

// ---------------------------------------------------------------------------
// Kernel 1: squared row norms d[b][i] = sum_k x[b][i][k]^2 ; zero row-sum buf
// ---------------------------------------------------------------------------
__global__ void bdc_init(const float* __restrict__ x,
                         float* __restrict__ d, float* __restrict__ s) {
    int idx = blockIdx.x * blockDim.x + threadIdx.x;
    if (idx >= NB * DIM) return;
    const float* row = x + (size_t)idx * MSP;
    float acc = 0.f;
    #pragma unroll 4
    for (int k = 0; k < MSP; ++k) acc += row[k] * row[k];
    d[idx] = acc;
    s[idx] = 0.f;
}

// ---------------------------------------------------------------------------
// Kernel 2: row sums of dcov over triu tiles only (dcov symmetric):
//   tile (I,J): row sums -> s[I-block rows]; col sums -> s[J-block rows] (J>I)
// ---------------------------------------------------------------------------
__global__ void bdc_rowsum(const float* __restrict__ x,
                           const float* __restrict__ d,
                           const float* __restrict__ t,
                           float* __restrict__ s) {
    __shared__ float smem[4 * 2 * 1600];        // per wave: A slot + B slot
    int b    = blockIdx.x / BLOCKS_PER_B;
    int wid  = threadIdx.x >> 5;
    int lane = threadIdx.x & 31;
    int tIdx = (blockIdx.x % BLOCKS_PER_B) * 4 + wid;
    int I, J; decode_triu(tIdx, I, J);

    const float* xb = x + (size_t)b * DIM * MSP;
    float* aslot = smem + wid * 3200;
    float* bslot = aslot + 1600;
    stage_tile(xb + I * 16 * MSP, aslot, lane);
    stage_tile(xb + J * 16 * MSP, bslot, lane);
    __syncthreads();

    int m16 = lane & 15, h = lane >> 4;
    v8f c = gram_tile_lds(aslot, bslot, m16, h);

    float expt = __expf(t[0]);
    const float* db = d + b * DIM;
    float dj = db[J * 16 + m16];                 // this lane's column norm
    float colv = 0.f;
    #pragma unroll
    for (int r = 0; r < 8; ++r) {
        float v = db[I * 16 + 8 * h + r] + dj - 2.0f * c[r];
        v = fmaxf(v, 0.0f);
        v = sqrtf(expt * v + EPS);
        colv += v;
        // reduce over the 16 columns held in this 16-lane half
        v += __shfl_xor(v, 1);
        v += __shfl_xor(v, 2);
        v += __shfl_xor(v, 4);
        v += __shfl_xor(v, 8);
        if (m16 == 0)
            atomicAdd(&s[b * DIM + I * 16 + 8 * h + r], v);
    }
    // symmetric contribution: tile column sums feed the J-block rows
    colv += __shfl_xor(colv, 16);
    if (h == 0 && J > I)
        atomicAdd(&s[b * DIM + J * 16 + m16], colv);
}

// ---------------------------------------------------------------------------
// Kernel 3: grand mean per batch: g[b] = sum_i s[b][i] / dim^2
// ---------------------------------------------------------------------------
__global__ void bdc_grand(const float* __restrict__ s, float* __restrict__ g) {
    int b = blockIdx.x;
    int tid = threadIdx.x;                       // 256 threads
    __shared__ float red[8];
    float acc = 0.f;
    for (int i = tid; i < DIM; i += 256) acc += s[b * DIM + i];
    for (int m = 1; m < 32; m <<= 1) acc += __shfl_xor(acc, m);
    if ((tid & 31) == 0) red[tid >> 5] = acc;
    __syncthreads();
    if (tid < 8) {
        float v = red[tid];
        for (int m = 1; m < 8; m <<= 1) v += __shfl_xor(v, m);
        if (tid == 0) g[b] = v * (INV_DIM * INV_DIM);
    }
}

// ---------------------------------------------------------------------------
// Kernel 4: recompute triu tiles, double-center, scatter to triu output.
// ---------------------------------------------------------------------------
__global__ void bdc_tri(const float* __restrict__ x,
                        const float* __restrict__ d,
                        const float* __restrict__ s,
                        const float* __restrict__ g,
                        const float* __restrict__ t,
                        float* __restrict__ out) {
    __shared__ float smem[4 * 2 * 1600];
    int b    = blockIdx.x / BLOCKS_PER_B;
    int wid  = threadIdx.x >> 5;
    int lane = threadIdx.x & 31;
    int tIdx = (blockIdx.x % BLOCKS_PER_B) * 4 + wid;
    int I, J; decode_triu(tIdx, I, J);

    const float* xb = x + (size_t)b * DIM * MSP;
    float* aslot = smem + wid * 3200;
    float* bslot = aslot + 1600;
    stage_tile(xb + I * 16 * MSP, aslot, lane);
    stage_tile(xb + J * 16 * MSP, bslot, lane);
    __syncthreads();

    int m16 = lane & 15, h = lane >> 4;
    v8f c = gram_tile_lds(aslot, bslot, m16, h);

    float expt = __expf(t[0]);
    float gb   = g[b];
    const float* db = d + b * DIM;
    const float* sb = s + b * DIM;

    int   gj = J * 16 + m16;
    float dj = db[gj];
    float sj = sb[gj] * INV_DIM;
    float di[8], si[8];
    #pragma unroll
    for (int r = 0; r < 8; ++r) {
        di[r] = db[I * 16 + 8 * h + r];
        si[r] = sb[I * 16 + 8 * h + r] * INV_DIM;
    }

    float* outb = out + (size_t)b * TRIU_PER_B;
    #pragma unroll
    for (int r = 0; r < 8; ++r) {
        int gi = I * 16 + 8 * h + r;
        float v = di[r] + dj - 2.0f * c[r];
        v = fmaxf(v, 0.0f);
        v = sqrtf(expt * v + EPS);
        float bdcv = v - si[r] - sj + gb;
        if (gj >= gi) {
            int idx = gi * DIM - (gi * (gi - 1)) / 2 + (gj - gi);
            outb[idx] = bdcv;
        }
    }
}

// ---------------------------------------------------------------------------
extern "C" void kernel_launch(void* const* d_in, const int* in_sizes, int n_in,
                              void* d_out, int out_size, void* d_ws, size_t ws_size,
                              hipStream_t stream) {
    const float* x = (const float*)d_in[0];
    const float* t = (const float*)d_in[1];
    float* out = (float*)d_out;

    float* d = (float*)d_ws;          // NB*DIM squared norms
    float* s = d + NB * DIM;          // NB*DIM row sums of dcov
    float* g = s + NB * DIM;          // NB grand means

    bdc_init<<<(NB * DIM + 255) / 256, 256, 0, stream>>>(x, d, s);
    bdc_rowsum<<<NB * BLOCKS_PER_B, 128, 0, stream>>>(x, d, t, s);
    bdc_grand<<<NB, 256, 0, stream>>>(s, g);
    bdc_tri<<<NB * BLOCKS_PER_B, 128, 0, stream>>>(x, d, s, g, t, out);
}